// AdaptiveRankTensorizedTextSubNet_67018669687329
// MI455X (gfx1250) — compile-verified
//
#include <hip/hip_runtime.h>

// AdaptiveRankTensorizedTextSubNet on MI455X (gfx1250, wave32).
// Pipeline:
//   prep_weights_kernel : zero-pad + transpose U_ih/U_hh -> UT[32][512] in ws
//                         (kills lane-divergent guard branches in GEMM loops,
//                          makes every B fragment one aligned b64 load)
//   xu_gemm_kernel      : xU = x @ U_ih  as one [81920x512]@[512x20] GEMM
//   lstm_recur_kernel   : persistent 16-row tiles, c-state in VGPRs,
//                         h-state in LDS, all matmuls V_WMMA_F32_16X16X4_F32.

typedef float v2f __attribute__((ext_vector_type(2)));
typedef float v8f __attribute__((ext_vector_type(8)));

#define NB    4096   // batch
#define NT    20     // time steps
#define DIN   512
#define HID   512
#define RNK   20     // CP rank
#define RPAD  32     // rank padded to 2 WMMA N-tiles
#define DOUT  64

// workspace layout (floats)
#define WS_XU     0                        // [NB*NT][RPAD]
#define WS_UTIH   (NB * NT * RPAD)         // [RPAD][DIN] transposed, zero-padded
#define WS_UTHH   (WS_UTIH + RPAD * DIN)   // [RPAD][HID] transposed, zero-padded

__device__ __forceinline__ v8f wmma_f32(v2f a, v2f b, v8f c) {
    // D = A(16x4) * B(4x16) + C(16x16), fp32 in/out
    return __builtin_amdgcn_wmma_f32_16x16x4_f32(
        false, a, false, b, (short)0, c, false, false);
}

__device__ __forceinline__ v2f ld2(const float* __restrict__ p) {
    return *(const v2f*)p;    // callers guarantee 8B alignment (even k)
}
__device__ __forceinline__ v2f ld2s(const float* p) {
    return *(const v2f*)p;    // LDS variant
}

__device__ __forceinline__ float fast_sigmoid(float x) {
    return 1.0f / (1.0f + __expf(-x));
}
__device__ __forceinline__ float fast_tanh(float x) {
    return 1.0f - 2.0f / (__expf(2.0f * x) + 1.0f);
}

// ---------------------------------------------------------------------------
// Prep: UT[n][k] = (n < RNK) ? U[k][n] : 0   for both U_ih and U_hh
// ---------------------------------------------------------------------------
__global__ __launch_bounds__(256)
void prep_weights_kernel(const float* __restrict__ U_ih,
                         const float* __restrict__ U_hh,
                         float* __restrict__ ut_ih,
                         float* __restrict__ ut_hh)
{
    const int i = blockIdx.x * 256 + threadIdx.x;   // [0, RPAD*DIN)
    if (i >= RPAD * DIN) return;
    const int n = i / DIN, k = i % DIN;
    const bool ok = (n < RNK);
    ut_ih[i] = ok ? U_ih[k * RNK + n] : 0.0f;
    ut_hh[i] = ok ? U_hh[k * RNK + n] : 0.0f;
}

// ---------------------------------------------------------------------------
// Kernel 1: xU[row, :] = x[row, :] @ U_ih, row in [0, B*T). One 16-row tile
// per wave, N = 32 (cols 20..31 are exact zeros via padded UT).
// ---------------------------------------------------------------------------
__global__ __launch_bounds__(256)
void xu_gemm_kernel(const float* __restrict__ x,      // [B*T, DIN]
                    const float* __restrict__ ut_ih,  // [RPAD, DIN]
                    float* __restrict__ xu)           // [B*T, RPAD]
{
    const int wave = threadIdx.x >> 5;
    const int lane = threadIdx.x & 31;
    const int lhi  = lane >> 4;
    const int llo  = lane & 15;
    const int row0 = (blockIdx.x * 8 + wave) * 16;
    if (row0 >= NB * NT) return;     // grid divides exactly; defensive

    const float* xrow = x + (size_t)(row0 + llo) * DIN;
    const float* b0p  = ut_ih + (size_t)llo * DIN;        // N tile 0
    const float* b1p  = ut_ih + (size_t)(16 + llo) * DIN; // N tile 1 (zeros>=20)

    v8f c0 = {}; v8f c1 = {};
#pragma unroll 4
    for (int k = 0; k < DIN; k += 4) {
        const int kk = k + 2 * lhi;
        const v2f a  = ld2(xrow + kk);
        const v2f b0 = ld2(b0p + kk);
        const v2f b1 = ld2(b1p + kk);
        c0 = wmma_f32(a, b0, c0);
        c1 = wmma_f32(a, b1, c1);
    }
    // C layout: (vgpr v, lane) -> M = v + 8*lhi, N = llo
#pragma unroll
    for (int v = 0; v < 8; ++v) {
        const int m = v + 8 * lhi;
        xu[(size_t)(row0 + m) * RPAD + llo]      = c0[v];
        xu[(size_t)(row0 + m) * RPAD + 16 + llo] = c1[v];
    }
}

// ---------------------------------------------------------------------------
// Kernel 2: LSTM recurrence + output projection. 16 batch rows per block,
// 8 waves. c-state in VGPRs, h-state in LDS, hU reduced via ds_add_f32.
// ---------------------------------------------------------------------------
__global__ __launch_bounds__(256)
void lstm_recur_kernel(const float* __restrict__ xu,    // [B*T, RPAD]
                       const float* __restrict__ V_ih,  // [4H, RNK]
                       const float* __restrict__ ut_hh, // [RPAD, HID]
                       const float* __restrict__ V_hh,  // [4H, RNK]
                       const float* __restrict__ W_out, // [DOUT, HID]
                       const float* __restrict__ b_out, // [DOUT]
                       float* __restrict__ out)         // [NB, DOUT]
{
    __shared__ float h_lds[16][HID];   // 32 KB
    __shared__ float hU[16][RPAD];     //  2 KB
    __shared__ float xU[16][RPAD];     //  2 KB
    __shared__ float yacc[16][DOUT];   //  4 KB

    const int tid   = threadIdx.x;
    const int wave  = tid >> 5;
    const int lane  = tid & 31;
    const int lhi   = lane >> 4;
    const int llo   = lane & 15;
    const int b0    = blockIdx.x * 16;
    const int jbase = wave * 64;       // this wave's hidden-column slice
    const int kbeg  = wave * 64;       // this wave's K slice for reductions

    for (int i = tid; i < 16 * HID; i += 256)
        (&h_lds[0][0])[i] = 0.0f;

    v8f creg[4];
#pragma unroll
    for (int jt = 0; jt < 4; ++jt) creg[jt] = (v8f){};

    for (int t = 0; t < NT; ++t) {
        __syncthreads();  // prev-step h visible; prev gates phase done
        // stage xU tile (one b64 per thread); zero hU accumulator
        {
            const int m = tid >> 4, n2 = (tid & 15) * 2;   // 256 thr x 2 floats
            *(v2f*)&xU[m][n2] =
                ld2(xu + ((size_t)(b0 + m) * NT + t) * RPAD + n2);
            *(v2f*)&hU[m][n2] = (v2f){};
        }
        // prefetch next step's xU tile into cache while we compute
        if (t + 1 < NT && tid < 32) {
            const int m = tid >> 1, half = (tid & 1) * 16;
            __builtin_prefetch(xu + ((size_t)(b0 + m) * NT + t + 1) * RPAD + half,
                               0, 1);
        }
        __syncthreads();

        // ---- hU = h @ U_hh (this wave: K in [kbeg, kbeg+64)) ----
        {
            const float* bb0p = ut_hh + (size_t)llo * HID;
            const float* bb1p = ut_hh + (size_t)(16 + llo) * HID;
            v8f d0 = {}, d1 = {};
#pragma unroll 4
            for (int ks = 0; ks < 16; ++ks) {
                const int k = kbeg + ks * 4 + 2 * lhi;
                const v2f a   = ld2s(&h_lds[llo][k]);
                const v2f bb0 = ld2(bb0p + k);
                const v2f bb1 = ld2(bb1p + k);
                d0 = wmma_f32(a, bb0, d0);
                d1 = wmma_f32(a, bb1, d1);
            }
#pragma unroll
            for (int v = 0; v < 8; ++v) {
                const int m = v + 8 * lhi;
                atomicAdd(&hU[m][llo],      d0[v]);
                atomicAdd(&hU[m][16 + llo], d1[v]);
            }
        }
        __syncthreads();

        // ---- preload A fragments (shared across all 16 gate tiles) ----
        v2f aX[5], aH[5];
#pragma unroll
        for (int s = 0; s < 5; ++s) {
            const int k = s * 4 + 2 * lhi;
            aX[s] = ld2s(&xU[llo][k]);
            aH[s] = ld2s(&hU[llo][k]);
        }

        // ---- gates + elementwise LSTM for this wave's 64 hidden cols ----
#pragma unroll
        for (int jt = 0; jt < 4; ++jt) {
            v8f g4[4];
#pragma unroll
            for (int q = 0; q < 4; ++q) {
                const int nrow = q * HID + jbase + jt * 16 + llo; // gate index
                const float* vi = V_ih + (size_t)nrow * RNK;
                const float* vh = V_hh + (size_t)nrow * RNK;
                v8f cc = {};
#pragma unroll
                for (int s = 0; s < 5; ++s) {
                    const int k = s * 4 + 2 * lhi;     // even -> aligned b64
                    cc = wmma_f32(aX[s], ld2(vi + k), cc);
                    cc = wmma_f32(aH[s], ld2(vh + k), cc);
                }
                g4[q] = cc;
            }
#pragma unroll
            for (int v = 0; v < 8; ++v) {
                const int m = v + 8 * lhi;
                const int j = jbase + jt * 16 + llo;
                const float ig = fast_sigmoid(g4[0][v]);
                const float fg = fast_sigmoid(g4[1][v]);
                const float gg = fast_tanh(g4[2][v]);
                const float og = fast_sigmoid(g4[3][v]);
                const float cv = fg * creg[jt][v] + ig * gg;
                creg[jt][v] = cv;
                h_lds[m][j] = og * fast_tanh(cv);
            }
        }
    }
    __syncthreads();

    // ---- y = h @ W_out^T + b_out ----
    for (int i = tid; i < 16 * DOUT; i += 256)
        yacc[i >> 6][i & 63] = b_out[i & 63];
    __syncthreads();
    {
        v8f dn[4];
#pragma unroll
        for (int nt = 0; nt < 4; ++nt) dn[nt] = (v8f){};
#pragma unroll 4
        for (int ks = 0; ks < 16; ++ks) {
            const int k = kbeg + ks * 4 + 2 * lhi;
            const v2f a = ld2s(&h_lds[llo][k]);
#pragma unroll
            for (int nt = 0; nt < 4; ++nt) {
                const int n = nt * 16 + llo;
                dn[nt] = wmma_f32(a, ld2(W_out + (size_t)n * HID + k), dn[nt]);
            }
        }
#pragma unroll
        for (int nt = 0; nt < 4; ++nt)
#pragma unroll
            for (int v = 0; v < 8; ++v)
                atomicAdd(&yacc[v + 8 * lhi][nt * 16 + llo], dn[nt][v]);
    }
    __syncthreads();
    for (int i = tid; i < 16 * DOUT; i += 256) {
        const int m = i >> 6, n = i & 63;
        out[(size_t)(b0 + m) * DOUT + n] = yacc[m][n];
    }
}

// ---------------------------------------------------------------------------
extern "C" void kernel_launch(void* const* d_in, const int* in_sizes, int n_in,
                              void* d_out, int out_size, void* d_ws, size_t ws_size,
                              hipStream_t stream) {
    (void)in_sizes; (void)n_in; (void)out_size; (void)ws_size;
    const float* x     = (const float*)d_in[0];
    const float* U_ih  = (const float*)d_in[1];
    const float* V_ih  = (const float*)d_in[2];
    const float* U_hh  = (const float*)d_in[3];
    const float* V_hh  = (const float*)d_in[4];
    const float* W_out = (const float*)d_in[5];
    const float* b_out = (const float*)d_in[6];
    float* out   = (float*)d_out;
    float* ws    = (float*)d_ws;   // ~10.6 MB used
    float* xuw   = ws + WS_XU;
    float* utih  = ws + WS_UTIH;
    float* uthh  = ws + WS_UTHH;

    prep_weights_kernel<<<(RPAD * DIN + 255) / 256, 256, 0, stream>>>(
        U_ih, U_hh, utih, uthh);
    // 81920 rows -> 5120 wave tiles -> 640 blocks of 8 waves
    xu_gemm_kernel<<<(NB * NT / 16) / 8, 256, 0, stream>>>(x, utih, xuw);
    // one block per 16 batch rows
    lstm_recur_kernel<<<NB / 16, 256, 0, stream>>>(xuw, V_ih, uthh, V_hh,
                                                   W_out, b_out, out);
}